// PSAMask_30554397344333
// MI455X (gfx1250) — compile-verified
//
#include <hip/hip_runtime.h>

// PSA-mask 'collect' for MI455X (gfx1250).
// out[b, hn*65+wn, h, w] = x[b, (hn-h+32)*65+(wn-w+32), h, w] if shifts in [0,65), else 0
// Pure data movement (0 FLOPs): HBM-bound, ~571 MB total -> ~25us floor at 23.3 TB/s.
// Strategy: 1 block per (b, hn, h); stage the 65x65 (dw,w) input slab in LDS via
// async global->LDS loads (coalesced rows), then emit coalesced 65x65 (wn,w) output
// rows with the w-shift resolved through a bank-conflict-free padded LDS read.

#define NH 65
#define NW 65
#define HMASK 65
#define WMASK 65
#define HALF 32
#define PLANE 4225        // 65*65
#define LDS_STRIDE 66     // pad: shifted read has lane-delta -65 -> bank-delta -1 (conflict free)
#define NTHREADS 256

#if defined(__has_builtin)
#if __has_builtin(__builtin_amdgcn_global_load_async_to_lds_b32)
#define HAVE_ASYNC_BUILTIN 1
#endif
#if __has_builtin(__builtin_amdgcn_s_wait_asynccnt)
#define HAVE_ASYNCCNT_BUILTIN 1
#endif
#endif

typedef __attribute__((address_space(1))) int g_int;
typedef __attribute__((address_space(3))) int l_int;

__device__ __forceinline__ void async_copy_b32(const float* gptr, float* lptr) {
#if defined(HAVE_ASYNC_BUILTIN)
    __builtin_amdgcn_global_load_async_to_lds_b32(
        (g_int*)gptr, (l_int*)lptr, /*offset=*/0, /*cpol=*/0);
#else
    // inline-asm fallback: vdst = LDS byte address (low 32 bits of generic shared ptr),
    // vaddr = 64-bit global address
    unsigned lds_off = (unsigned)(unsigned long long)lptr;
    asm volatile("global_load_async_to_lds_b32 %0, %1, off"
                 :: "v"(lds_off), "v"(gptr) : "memory");
#endif
}

__device__ __forceinline__ void async_wait_all() {
#if defined(HAVE_ASYNCCNT_BUILTIN)
    __builtin_amdgcn_s_wait_asynccnt(0);
#else
    asm volatile("s_wait_asynccnt 0" ::: "memory");
#endif
}

__global__ __launch_bounds__(NTHREADS)
void psamask_collect_kernel(const float* __restrict__ x, float* __restrict__ out) {
    const int h   = blockIdx.x;   // 0..64
    const int hn  = blockIdx.y;   // 0..64
    const int b   = blockIdx.z;   // 0..3
    const int tid = threadIdx.x;

    // out[b, hn*65+wn, h, w] ; base covers wn=0,w=0, add wn*PLANE + w
    float* outp = out + ((size_t)(b * PLANE + hn * NW) * PLANE + h * NW);

    const int dh = hn - h + HALF;
    if ((unsigned)dh >= (unsigned)HMASK) {
        // whole (wn,w) plane is invalid -> stream zeros (NT: output never re-read)
        for (int i = tid; i < PLANE; i += NTHREADS) {
            int wn = i / NW;
            int w  = i - wn * NW;
            __builtin_nontemporal_store(0.0f, outp + wn * PLANE + w);
        }
        return;
    }

    __shared__ float tile[HMASK * LDS_STRIDE];   // 65*66*4 = 17160 B

    // x[b, dh*65+dw, h, w] ; base covers dw=0,w=0, add dw*PLANE + w
    const float* inp = x + ((size_t)(b * PLANE + dh * WMASK) * PLANE + h * NW);

    // Stage input slab: 65 rows (dw) of 65 contiguous floats -> coalesced,
    // pushed straight into LDS via the CDNA5 async path (ASYNCcnt).
    for (int i = tid; i < PLANE; i += NTHREADS) {
        int dw = i / NW;
        int w  = i - dw * NW;
        async_copy_b32(inp + dw * PLANE + w, &tile[dw * LDS_STRIDE + w]);
    }
    async_wait_all();      // my wave's async loads landed in LDS
    __syncthreads();       // everyone's did

    // Emit output: for each wn, a contiguous 65-float row (coalesced NT stores).
    // LDS read index dw*66 + w with dw = wn - w + 32: lane delta -65 -> no bank conflicts.
    for (int i = tid; i < PLANE; i += NTHREADS) {
        int wn = i / NW;
        int w  = i - wn * NW;
        int dw = wn - w + HALF;
        float v = ((unsigned)dw < (unsigned)WMASK) ? tile[dw * LDS_STRIDE + w] : 0.0f;
        __builtin_nontemporal_store(v, outp + wn * PLANE + w);
    }
}

extern "C" void kernel_launch(void* const* d_in, const int* in_sizes, int n_in,
                              void* d_out, int out_size, void* d_ws, size_t ws_size,
                              hipStream_t stream) {
    const float* x  = (const float*)d_in[0];
    float* out      = (float*)d_out;
    dim3 grid(NH, HMASK, 4);     // (h, hn, b) = 65*65*4 = 16900 blocks
    psamask_collect_kernel<<<grid, NTHREADS, 0, stream>>>(x, out);
}